// TDConv_49478023250588
// MI455X (gfx1250) — compile-verified
//
#include <hip/hip_runtime.h>

// Problem dims (fixed by the reference)
#define B_ 8
#define C_ 64
#define N_ 4096
#define O_ 64
#define K_ 16
#define CH 512   // m-chunk staged in LDS per top-K merge round

typedef float v2f __attribute__((ext_vector_type(2)));
typedef float v8f __attribute__((ext_vector_type(8)));

// ---------------------------------------------------------------------------
// 1) sq[b*N+n] = sum_c x[b][c][n]^2
// ---------------------------------------------------------------------------
__global__ __launch_bounds__(256) void sq_kernel(const float* __restrict__ x,
                                                 float* __restrict__ sq) {
  int i = blockIdx.x * 256 + threadIdx.x;       // [0, B*N)
  int b = i >> 12;
  int n = i & (N_ - 1);
  const float* xb = x + (size_t)b * C_ * N_;
  float s = 0.f;
#pragma unroll
  for (int c = 0; c < C_; ++c) {
    float v = xb[c * N_ + n];
    s += v * v;
  }
  sq[i] = s;
}

// ---------------------------------------------------------------------------
// 2) KNN: per (b, 16-row strip) compute neg_dist tile-by-tile with
//    V_WMMA_F32_16X16X4_F32 (exact fp32 path). Loads are decoupled from the
//    WMMA chain (load-all-fragments, then 16 back-to-back WMMAs) and the
//    8-tile chunk loop is fully unrolled so the scheduler can prefetch tile
//    t+1's B-fragments behind tile t's WMMA chain. Distances are staged in
//    LDS (padded rows), and lanes 0..15 maintain running sorted top-16 lists.
//    idx holds GLOBAL row ids (b*N + m) to simplify later gathers.
// ---------------------------------------------------------------------------
__global__ __launch_bounds__(128) void knn_kernel(const float* __restrict__ x,
                                                  const float* __restrict__ sq,
                                                  int* __restrict__ idx) {
  __shared__ float distS[16][CH + 1];   // +1 pad: row stride 513 words -> no bank conflict
  __shared__ float sqrow[16];

  const int tid  = threadIdx.x;
  const int lane = tid & 31;
  const int wave = tid >> 5;            // 4 waves
  const int col  = lane & 15;
  const int koff = (lane < 16) ? 0 : 2; // fp32 16x4 A / 4x16 B lane->K mapping
  const int b    = blockIdx.y;
  const int n0   = blockIdx.x * 16;

  const float* xb  = x + (size_t)b * C_ * N_;
  const float* sqb = sq + b * N_;

  // Hoist A fragments (16 query rows, all C=64) -> 16 x v2f, reused for all m.
  v2f afrag[16];
#pragma unroll
  for (int kk = 0; kk < 16; ++kk) {
    int k = 4 * kk + koff;
    afrag[kk].x = xb[k * N_ + n0 + col];
    afrag[kk].y = xb[(k + 1) * N_ + n0 + col];
  }
  if (tid < 16) sqrow[tid] = sqb[n0 + tid];

  // Running top-K (sorted descending) in lanes 0..15 of wave 0.
  float tv[K_];
  int   ti[K_];
#pragma unroll
  for (int k = 0; k < K_; ++k) { tv[k] = -__builtin_inff(); ti[k] = 0; }

  __syncthreads();

  for (int chunk = 0; chunk < N_; chunk += CH) {
    // --- compute phase: 4 waves cover CH/16 = 32 tiles of 16 columns.
    //     Fully unrolled so loads of the next tile overlap this tile's WMMAs.
#pragma unroll
    for (int t = 0; t < CH / 64; ++t) {
      const int m0 = chunk + (wave + 4 * t) * 16;

      // Load phase: all 16 B-fragments first (one clause, one wait).
      v2f bfrag[16];
#pragma unroll
      for (int kk = 0; kk < 16; ++kk) {
        int k = 4 * kk + koff;
        bfrag[kk].x = xb[k * N_ + m0 + col];
        bfrag[kk].y = xb[(k + 1) * N_ + m0 + col];
      }
      float sqm = sqb[m0 + col];

      // Compute phase: 16 back-to-back WMMAs chained through the accumulator.
      v8f acc = {};
#pragma unroll
      for (int kk = 0; kk < 16; ++kk) {
        acc = __builtin_amdgcn_wmma_f32_16x16x4_f32(
            false, afrag[kk], false, bfrag[kk], (short)0, acc, false, false);
      }

      const int rbase = (lane < 16) ? 0 : 8;
      const int mloc  = m0 - chunk + col;
#pragma unroll
      for (int v = 0; v < 8; ++v)
        distS[rbase + v][mloc] = 2.f * acc[v] - sqrow[rbase + v] - sqm;
    }
    __syncthreads();

    // --- merge phase: lane r scans its row's CH candidates ---
    if (tid < 16) {
      for (int j = 0; j < CH; ++j) {
        float v = distS[tid][j];
        if (v > tv[K_ - 1]) {
          int p = K_ - 1;
          while (p > 0 && tv[p - 1] < v) {
            tv[p] = tv[p - 1]; ti[p] = ti[p - 1]; --p;
          }
          tv[p] = v;
          ti[p] = b * N_ + chunk + j;    // global row id
        }
      }
    }
    __syncthreads();
  }

  if (tid < 16) {
    int* op = idx + (size_t)(b * N_ + n0 + tid) * K_;
#pragma unroll
    for (int k = 0; k < K_; ++k) op[k] = ti[k];
  }
}

// ---------------------------------------------------------------------------
// 3) U = Xt * W1^T, V = Xt * (W2 - W1)^T   (both via f32 WMMA)
//    Then y[b,n,k,o] = U[idx[b,n,k]][o] + V[b*N+n][o]  (never materialized)
// ---------------------------------------------------------------------------
__global__ __launch_bounds__(128) void uv_kernel(const float* __restrict__ x,
                                                 const float* __restrict__ W,
                                                 float* __restrict__ U,
                                                 float* __restrict__ V) {
  const int tid   = threadIdx.x;
  const int lane  = tid & 31;
  const int otile = tid >> 5;           // wave -> 16-wide o tile
  const int col   = lane & 15;
  const int koff  = (lane < 16) ? 0 : 2;
  const int p0    = blockIdx.x * 16;    // global row over B*N (never straddles b)
  const int b     = p0 >> 12;
  const int n0    = p0 & (N_ - 1);
  const float* xb = x + (size_t)b * C_ * N_;

  // Load phase: A fragments and both W-derived B fragment sets up front.
  v2f afrag[16];
#pragma unroll
  for (int kk = 0; kk < 16; ++kk) {
    int k = 4 * kk + koff;
    afrag[kk].x = xb[k * N_ + n0 + col];
    afrag[kk].y = xb[(k + 1) * N_ + n0 + col];
  }

  const int o = otile * 16 + col;
  const float* wo = W + (size_t)o * 2 * C_;

  v2f bu[16], bv[16];
#pragma unroll
  for (int kk = 0; kk < 16; ++kk) {
    int k = 4 * kk + koff;
    float w0 = wo[k], w1 = wo[k + 1];
    float w2 = wo[C_ + k], w3 = wo[C_ + k + 1];
    bu[kk].x = w0;       bu[kk].y = w1;
    bv[kk].x = w2 - w0;  bv[kk].y = w3 - w1;
  }

  // Compute phase: two independent 16-WMMA accumulation chains.
  v8f accU = {};
  v8f accV = {};
#pragma unroll
  for (int kk = 0; kk < 16; ++kk) {
    accU = __builtin_amdgcn_wmma_f32_16x16x4_f32(
        false, afrag[kk], false, bu[kk], (short)0, accU, false, false);
    accV = __builtin_amdgcn_wmma_f32_16x16x4_f32(
        false, afrag[kk], false, bv[kk], (short)0, accV, false, false);
  }
  const int rbase = (lane < 16) ? 0 : 8;
#pragma unroll
  for (int v = 0; v < 8; ++v) {
    U[(size_t)(p0 + rbase + v) * O_ + o] = accU[v];
    V[(size_t)(p0 + rbase + v) * O_ + o] = accV[v];
  }
}

// ---------------------------------------------------------------------------
// 4) BN statistics: sum & sumsq per channel via LDS partials + f32 atomics.
// ---------------------------------------------------------------------------
__global__ void zero_kernel(float* __restrict__ stats) {
  if (threadIdx.x < 128) stats[threadIdx.x] = 0.f;
}

__global__ __launch_bounds__(256) void stats_kernel(const float* __restrict__ U,
                                                    const float* __restrict__ V,
                                                    const int* __restrict__ idx,
                                                    float* __restrict__ stats) {
  __shared__ float r1[4][64], r2[4][64];
  const int o = threadIdx.x & 63;
  const int g = threadIdx.x >> 6;
  const int PPB = (B_ * N_) / 128;      // grid = 128 blocks
  const int p0  = blockIdx.x * PPB;
  float s1 = 0.f, s2 = 0.f;
  for (int p = p0 + g; p < p0 + PPB; p += 4) {
    float vv = V[(size_t)p * O_ + o];
    const int* ip = idx + (size_t)p * K_;
#pragma unroll
    for (int k = 0; k < K_; ++k) {
      float y = U[(size_t)ip[k] * O_ + o] + vv;
      s1 += y;
      s2 += y * y;
    }
  }
  r1[g][o] = s1;
  r2[g][o] = s2;
  __syncthreads();
  if (g == 0) {
    s1 = r1[0][o] + r1[1][o] + r1[2][o] + r1[3][o];
    s2 = r2[0][o] + r2[1][o] + r2[2][o] + r2[3][o];
    atomicAdd(&stats[o], s1);
    atomicAdd(&stats[64 + o], s2);
  }
}

// ---------------------------------------------------------------------------
// 5) Recompute y on the fly, BN + leaky-ReLU, max over k, write (B,O,N).
// ---------------------------------------------------------------------------
__global__ __launch_bounds__(256) void final_kernel(const float* __restrict__ U,
                                                    const float* __restrict__ V,
                                                    const int* __restrict__ idx,
                                                    const float* __restrict__ stats,
                                                    const float* __restrict__ gamma,
                                                    const float* __restrict__ beta,
                                                    float* __restrict__ out) {
  const int o = threadIdx.x & 63;
  const int g = threadIdx.x >> 6;
  const int p = blockIdx.x * 4 + g;     // global (b,n) pair
  const float invM = 1.f / (float)(B_ * N_ * K_);
  float mean = stats[o] * invM;
  float var  = stats[64 + o] * invM - mean * mean;
  float rs   = rsqrtf(var + 1e-5f);
  float ga = gamma[o], be = beta[o];
  float vv = V[(size_t)p * O_ + o];
  const int* ip = idx + (size_t)p * K_;
  float best = -__builtin_inff();
#pragma unroll
  for (int k = 0; k < K_; ++k) {
    float y  = U[(size_t)ip[k] * O_ + o] + vv;
    float yn = (y - mean) * rs * ga + be;
    float a  = yn >= 0.f ? yn : 0.2f * yn;
    best = fmaxf(best, a);
  }
  const int b = p >> 12;
  const int n = p & (N_ - 1);
  out[(size_t)b * O_ * N_ + (size_t)o * N_ + n] = best;
}

// ---------------------------------------------------------------------------
extern "C" void kernel_launch(void* const* d_in, const int* in_sizes, int n_in,
                              void* d_out, int out_size, void* d_ws, size_t ws_size,
                              hipStream_t stream) {
  const float* x     = (const float*)d_in[0];   // (8,64,4096)
  const float* W     = (const float*)d_in[1];   // (64,128)
  const float* gamma = (const float*)d_in[2];   // (64,)
  const float* beta  = (const float*)d_in[3];   // (64,)
  float* out = (float*)d_out;                   // (8,64,4096)

  // Workspace layout (~19 MB total):
  float* ws    = (float*)d_ws;
  float* sq    = ws;                              // B*N            = 32768 f
  float* U     = sq + B_ * N_;                    // B*N*O          = 2097152 f
  float* V     = U + (size_t)B_ * N_ * O_;        // B*N*O          = 2097152 f
  float* stats = V + (size_t)B_ * N_ * O_;        // 128 f
  int*   idx   = (int*)(stats + 128);             // B*N*K          = 524288 i

  sq_kernel<<<(B_ * N_) / 256, 256, 0, stream>>>(x, sq);
  knn_kernel<<<dim3(N_ / 16, B_), 128, 0, stream>>>(x, sq, idx);
  uv_kernel<<<(B_ * N_) / 16, 128, 0, stream>>>(x, W, U, V);
  zero_kernel<<<1, 128, 0, stream>>>(stats);
  stats_kernel<<<128, 256, 0, stream>>>(U, V, idx, stats);
  final_kernel<<<(B_ * N_) / 4, 256, 0, stream>>>(U, V, idx, stats, gamma, beta, out);
}